// EnhancedFlowGNN_71889162600747
// MI455X (gfx1250) — compile-verified
//
#include <hip/hip_runtime.h>
#include <cstdint>

#define HDIM 128
#define DIN  18

typedef __attribute__((ext_vector_type(2))) float v2f;
typedef __attribute__((ext_vector_type(8))) float v8f;

__device__ __forceinline__ float leaky_f(float v) { return v >= 0.f ? v : 0.2f * v; }

__device__ __forceinline__ void atomicMaxF(float* addr, float val) {
    int* iaddr = (int*)addr;
    int cur = *iaddr;
    while (__int_as_float(cur) < val) {
        int prev = atomicCAS(iaddr, cur, __float_as_int(val));
        if (prev == cur) break;
        cur = prev;
    }
}

// ---------------- utility fill ----------------
__global__ void fill_kernel(float* __restrict__ p, float val, long long n) {
    long long i = (long long)blockIdx.x * blockDim.x + threadIdx.x;
    if (i < n) p[i] = val;
}

// ---------------- neighbor sum + degree ----------------
__global__ void edge_deg_nsum(const int* __restrict__ row, const int* __restrict__ col,
                              const float* __restrict__ x, float* __restrict__ nsum,
                              float* __restrict__ deg, int E) {
    int t = blockIdx.x * blockDim.x + threadIdx.x;
    if (t >= E) return;
    int r = row[t], c = col[t];
    atomicAdd(&deg[r], 1.0f);
    #pragma unroll
    for (int d = 0; d < DIN; ++d)
        atomicAdd(&nsum[(long long)r * DIN + d], x[(long long)c * DIN + d]);
}

// ---------------- input projection: h0 = x@W_in + nmean@W_agg + b ----------------
__global__ __launch_bounds__(128) void input_proj(const float* __restrict__ x,
        const float* __restrict__ nsum, const float* __restrict__ deg,
        const float* __restrict__ W_in, const float* __restrict__ b_in,
        const float* __restrict__ W_agg, const float* __restrict__ b_agg,
        float* __restrict__ H0, int M) {
    int node = blockIdx.x;
    if (node >= M) return;
    __shared__ float sx[DIN], sn[DIN];
    if (threadIdx.x < DIN) {
        sx[threadIdx.x] = x[(long long)node * DIN + threadIdx.x];
        float dg = deg[node];
        sn[threadIdx.x] = nsum[(long long)node * DIN + threadIdx.x] / (dg + 1e-8f);
    }
    __syncthreads();
    int o = threadIdx.x;
    float acc = b_in[o] + b_agg[o];
    #pragma unroll
    for (int d = 0; d < DIN; ++d)
        acc += sx[d] * W_in[d * HDIM + o] + sn[d] * W_agg[d * HDIM + o];
    H0[(long long)node * HDIM + o] = acc;
}

// ---------------- WMMA f32 GEMM: Y[M,128] = X[M,128] @ W[128,128] (+bias, opt leaky) ---
// A frag (16x4 f32): lanes 0-15 hold row M=lane, K={k,k+1}; lanes 16-31 K={k+2,k+3}
// B frag (4x16 f32): lanes 0-15 hold col N=lane, K={k,k+1}; lanes 16-31 K={k+2,k+3}
// C/D (16x16 f32): VGPR r -> (M=r, N=lane) for lanes 0-15; (M=r+8, N=lane-16) else
__global__ __launch_bounds__(256) void gemm_h128(
        const float* __restrict__ X, const float* __restrict__ W,
        const float* __restrict__ bias, float* __restrict__ Y, int M, int act) {
    __shared__ float sW[HDIM * HDIM];           // 64 KB of 320 KB WGP LDS
    {
        const float4* W4 = (const float4*)W;
        float4* s4 = (float4*)sW;
        for (int i = threadIdx.x; i < HDIM * HDIM / 4; i += 256) s4[i] = W4[i];
    }
    __syncthreads();

    const int lane = threadIdx.x & 31;
    const int wave = threadIdx.x >> 5;
    const int rowTile = blockIdx.x * 128 + wave * 16;   // 16 rows per wave
    const int m = lane & 15;
    const int khalf = lane >> 4;

    int row = rowTile + m;
    int rowC = row < M ? row : (M - 1);                 // clamp for safe loads
    const float* __restrict__ xrow = X + (long long)rowC * HDIM;

    v8f acc[8] = {};

    for (int kk = 0; kk < HDIM; kk += 4) {
        const int kb = kk + 2 * khalf;
        v2f a;
        a.x = xrow[kb];
        a.y = xrow[kb + 1];
        #pragma unroll
        for (int t = 0; t < 8; ++t) {
            const int n = t * 16 + m;
            v2f b;
            b.x = sW[kb * HDIM + n];
            b.y = sW[(kb + 1) * HDIM + n];
            acc[t] = __builtin_amdgcn_wmma_f32_16x16x4_f32(
                false, a, false, b, (short)0, acc[t], false, false);
        }
    }

    const int rbase = khalf * 8;
    #pragma unroll
    for (int t = 0; t < 8; ++t) {
        const int n = t * 16 + m;
        const float bv = bias ? bias[n] : 0.f;
        #pragma unroll
        for (int r = 0; r < 8; ++r) {
            const int gr = rowTile + rbase + r;
            if (gr < M) {
                float y = acc[t][r] + bv;
                if (act) y = leaky_f(y);
                Y[(long long)gr * HDIM + n] = y;
            }
        }
    }
}

// ---------------- per-node attention scalars: S[n,v] = H[n,:] . A[v,:] ----------------
__global__ void dots_kernel(const float* __restrict__ Hm, const float* __restrict__ A,
                            float* __restrict__ S, int M, int nvec) {
    int node = blockIdx.x * (blockDim.x >> 5) + (threadIdx.x >> 5);
    int lane = threadIdx.x & 31;
    if (node >= M) return;
    float4 hv = ((const float4*)(Hm + (long long)node * HDIM))[lane];
    for (int v = 0; v < nvec; ++v) {
        float4 av = ((const float4*)(A + v * HDIM))[lane];
        float p = hv.x * av.x + hv.y * av.y + hv.z * av.z + hv.w * av.w;
        for (int off = 16; off > 0; off >>= 1) p += __shfl_xor(p, off, 32);
        if (lane == 0) S[(long long)node * nvec + v] = p;
    }
}

// ---------------- edge pass 1: e = leaky(ssrc[row]+sdst[col]); segment max ------------
__global__ void edge_logits(const int* __restrict__ row, const int* __restrict__ col,
                            const float* __restrict__ ssrc, const float* __restrict__ sdst,
                            float* __restrict__ eb, float* __restrict__ mx, int E, int nh) {
    int t = blockIdx.x * blockDim.x + threadIdx.x;
    if (t >= E * nh) return;
    int e = t / nh, hh = t - e * nh;
    int r = row[e], c = col[e];
    float v = leaky_f(ssrc[(long long)r * nh + hh] + sdst[(long long)c * nh + hh]);
    eb[t] = v;
    atomicMaxF(&mx[(long long)r * nh + hh], v);
}

// ---------------- edge pass 2: ex = exp(e - max); segment sum --------------------------
__global__ void edge_exp(const int* __restrict__ row, float* __restrict__ eb,
                         const float* __restrict__ mx, float* __restrict__ ssum,
                         int E, int nh) {
    int t = blockIdx.x * blockDim.x + threadIdx.x;
    if (t >= E * nh) return;
    int e = t / nh, hh = t - e * nh;
    int r = row[e];
    float ex = expf(eb[t] - mx[(long long)r * nh + hh]);
    eb[t] = ex;
    atomicAdd(&ssum[(long long)r * nh + hh], ex);
}

// ---------------- edge pass 3: O[row] += alpha * V[col] (4 feats / thread) ------------
__global__ void edge_scatter(const int* __restrict__ row, const int* __restrict__ col,
                             const float* __restrict__ eb, const float* __restrict__ ssum,
                             const float* __restrict__ V, float* __restrict__ O,
                             int E, int nh) {
    long long t = (long long)blockIdx.x * blockDim.x + threadIdx.x;
    if (t >= (long long)E * 32) return;
    int e = (int)(t >> 5);
    int g = (int)(t & 31);
    int f = g * 4;
    int hh = (g * nh) >> 5;                       // head owning feature f (DH=32)
    int r = row[e], c = col[e];
    float alpha = eb[(long long)e * nh + hh] /
                  (ssum[(long long)r * nh + hh] + 1e-16f);
    float4 v4 = *(const float4*)(V + (long long)c * HDIM + f);
    float* o = O + (long long)r * HDIM + f;
    atomicAdd(o + 0, alpha * v4.x);
    atomicAdd(o + 1, alpha * v4.y);
    atomicAdd(o + 2, alpha * v4.z);
    atomicAdd(o + 3, alpha * v4.w);
}

// ---------------- BN(eval) + optional residual + leaky --------------------------------
__global__ void bn_res_leaky(const float* __restrict__ Hin, const float* __restrict__ Res,
                             const float* __restrict__ g, const float* __restrict__ bt,
                             const float* __restrict__ mn, const float* __restrict__ vr,
                             float* __restrict__ Hout, int M) {
    long long t = (long long)blockIdx.x * blockDim.x + threadIdx.x;
    if (t >= (long long)M * HDIM) return;
    int fcol = (int)(t & (HDIM - 1));
    float y = (Hin[t] - mn[fcol]) * rsqrtf(vr[fcol] + 1e-5f) * g[fcol] + bt[fcol];
    if (Res) y += Res[t];
    Hout[t] = leaky_f(y);
}

// ---------------- repack mh_Wv [4,128,32] -> [128,128] (d -> h*32+e) -------------------
__global__ void repack_mhW(const float* __restrict__ Wm, float* __restrict__ Wp) {
    int i = blockIdx.x * blockDim.x + threadIdx.x;
    if (i >= HDIM * HDIM) return;
    int d = i >> 7, o = i & 127;
    int hh = o >> 5, e = o & 31;
    Wp[i] = Wm[hh * HDIM * 32 + d * 32 + e];
}

// ---------------- output head: out = x[:,15:18] + T@W2 + b2 --------------------------
__global__ void out_head(const float* __restrict__ T, const float* __restrict__ W2,
                         const float* __restrict__ b2, const float* __restrict__ x,
                         float* __restrict__ out, int M) {
    int node = blockIdx.x * (blockDim.x >> 5) + (threadIdx.x >> 5);
    int lane = threadIdx.x & 31;
    if (node >= M) return;
    float4 tv = ((const float4*)(T + (long long)node * HDIM))[lane];
    #pragma unroll
    for (int j = 0; j < 3; ++j) {
        float p = tv.x * W2[(lane * 4 + 0) * 3 + j] + tv.y * W2[(lane * 4 + 1) * 3 + j] +
                  tv.z * W2[(lane * 4 + 2) * 3 + j] + tv.w * W2[(lane * 4 + 3) * 3 + j];
        for (int off = 16; off > 0; off >>= 1) p += __shfl_xor(p, off, 32);
        if (lane == 0)
            out[(long long)node * 3 + j] = x[(long long)node * DIN + 15 + j] + b2[j] + p;
    }
}

extern "C" void kernel_launch(void* const* d_in, const int* in_sizes, int n_in,
                              void* d_out, int out_size, void* d_ws, size_t ws_size,
                              hipStream_t stream) {
    (void)n_in; (void)out_size; (void)ws_size;
    const float* x       = (const float*)d_in[0];
    const int*   ei      = (const int*)  d_in[1];
    const float* W_in    = (const float*)d_in[2];
    const float* b_in    = (const float*)d_in[3];
    const float* W_agg   = (const float*)d_in[4];
    const float* b_agg   = (const float*)d_in[5];
    const float* sh_Wv   = (const float*)d_in[6];
    const float* sh_b    = (const float*)d_in[7];
    const float* sh_asrc = (const float*)d_in[8];
    const float* sh_adst = (const float*)d_in[9];
    const float* mh_Wv   = (const float*)d_in[10];
    const float* mh_b    = (const float*)d_in[11];
    const float* mh_asrc = (const float*)d_in[12];
    const float* mh_adst = (const float*)d_in[13];
    const float* bn_g    = (const float*)d_in[14];
    const float* bn_b    = (const float*)d_in[15];
    const float* bn_m    = (const float*)d_in[16];
    const float* bn_v    = (const float*)d_in[17];
    const float* W_o1    = (const float*)d_in[18];
    const float* b_o1    = (const float*)d_in[19];
    const float* W_o2    = (const float*)d_in[20];
    const float* b_o2    = (const float*)d_in[21];

    const int N = in_sizes[0] / DIN;
    const int E = in_sizes[1] / 2;
    const int* rowp = ei;
    const int* colp = ei + E;

    float* ws   = (float*)d_ws;
    float* nsum = ws; ws += (long long)N * DIN;
    float* deg  = ws; ws += N;
    float* bufA = ws; ws += (long long)N * HDIM;
    float* bufB = ws; ws += (long long)N * HDIM;
    float* bufC = ws; ws += (long long)N * HDIM;
    float* ssrc = ws; ws += (long long)N * 4;
    float* sdst = ws; ws += (long long)N * 4;
    float* mbuf = ws; ws += (long long)N * 4;
    float* sbuf = ws; ws += (long long)N * 4;
    float* ebuf = ws; ws += (long long)E * 4;
    float* wpk  = ws; ws += HDIM * HDIM;

    auto fills = [&](float* p, float v, long long n) {
        fill_kernel<<<(int)((n + 255) / 256), 256, 0, stream>>>(p, v, n);
    };

    // 1) neighbor mean aggregation
    fills(nsum, 0.f, (long long)N * DIN);
    fills(deg, 0.f, N);
    edge_deg_nsum<<<(E + 255) / 256, 256, 0, stream>>>(rowp, colp, x, nsum, deg, E);
    // 2) input projection -> bufA (= x0)
    input_proj<<<N, 128, 0, stream>>>(x, nsum, deg, W_in, b_in, W_agg, b_agg, bufA, N);

    const int gemmGrid = (N + 127) / 128;
    const int waveGrid = (N + 7) / 8;
    const long long nh128 = (long long)N * HDIM;

    auto attn_layer = [&](const float* Hin, const float* Wv, const float* bv,
                          const float* asrc, const float* adst, int nh,
                          const float* res, int bnL, float* Vb, float* Ob, float* Hout) {
        gemm_h128<<<gemmGrid, 256, 0, stream>>>(Hin, Wv, bv, Vb, N, 0);
        dots_kernel<<<waveGrid, 256, 0, stream>>>(Hin, asrc, ssrc, N, nh);
        dots_kernel<<<waveGrid, 256, 0, stream>>>(Hin, adst, sdst, N, nh);
        fills(mbuf, -1e30f, (long long)N * nh);
        fills(sbuf, 0.f, (long long)N * nh);
        fills(Ob, 0.f, nh128);
        int tE = E * nh;
        edge_logits<<<(tE + 255) / 256, 256, 0, stream>>>(rowp, colp, ssrc, sdst, ebuf, mbuf, E, nh);
        edge_exp<<<(tE + 255) / 256, 256, 0, stream>>>(rowp, ebuf, mbuf, sbuf, E, nh);
        long long st = (long long)E * 32;
        edge_scatter<<<(int)((st + 255) / 256), 256, 0, stream>>>(rowp, colp, ebuf, sbuf, Vb, Ob, E, nh);
        bn_res_leaky<<<(int)((nh128 + 255) / 256), 256, 0, stream>>>(
            Ob, res, bn_g + bnL * HDIM, bn_b + bnL * HDIM,
            bn_m + bnL * HDIM, bn_v + bnL * HDIM, Hout, N);
    };

    // layer 0: single-head, residual = x0 (bufA)  -> h in bufC
    attn_layer(bufA, sh_Wv, sh_b, sh_asrc, sh_adst, 1, bufA, 0, bufB, bufC, bufC);
    // layer 1: multi-head (4 heads), no residual   -> h in bufA
    repack_mhW<<<(HDIM * HDIM + 255) / 256, 256, 0, stream>>>(mh_Wv, wpk);
    attn_layer(bufC, wpk, mh_b, mh_asrc, mh_adst, 4, nullptr, 1, bufB, bufA, bufA);
    // layer 2: single-head, residual = h           -> h in bufC
    attn_layer(bufA, sh_Wv + HDIM * HDIM, sh_b + HDIM, sh_asrc + HDIM, sh_adst + HDIM,
               1, bufA, 2, bufB, bufC, bufC);

    // output head: T = leaky(h @ W_o1 + b_o1); out = x[:,15:18] + T @ W_o2 + b_o2
    gemm_h128<<<gemmGrid, 256, 0, stream>>>(bufC, W_o1, b_o1, bufB, N, 1);
    out_head<<<waveGrid, 256, 0, stream>>>(bufB, W_o2, b_o2, x, (float*)d_out, N);
}